// TemporalLSTM_72928544686308
// MI455X (gfx1250) — compile-verified
//
#include <hip/hip_runtime.h>
#include <hip/hip_bf16.h>
#include <math.h>

typedef float v2f __attribute__((ext_vector_type(2)));
typedef float v8f __attribute__((ext_vector_type(8)));

constexpr int B  = 512;
constexpr int S  = 128;    // seq len (= attention width)
constexpr int D  = 128;    // feature dim
constexpr int HS = 256;    // hidden size
constexpr int G4 = 4 * HS; // 1024 gate columns
constexpr int KA = D + HS;      // 384: concatenated K for the step GEMM
constexpr int ASTRIDE = KA + 4; // 388: LDS row stride -> bank = (4*row + k) % 64

// --- gfx1250 async global->LDS staging (ASYNCcnt path), with safe fallback ---
#if defined(__has_builtin)
#  if __has_builtin(__builtin_amdgcn_global_load_async_to_lds_b128)
#    define USE_ASYNC_LDS 1
#  endif
#endif
#ifndef USE_ASYNC_LDS
#  define USE_ASYNC_LDS 0
#endif

#if USE_ASYNC_LDS
typedef int v4i_t __attribute__((vector_size(16)));
typedef __attribute__((address_space(1))) v4i_t* g_v4i_p;  // global v4i*
typedef __attribute__((address_space(3))) v4i_t* l_v4i_p;  // LDS v4i*
#endif

__device__ __forceinline__ void stage16(const float* __restrict__ src, float* dst) {
#if USE_ASYNC_LDS
  __builtin_amdgcn_global_load_async_to_lds_b128((g_v4i_p)src, (l_v4i_p)dst, 0, 0);
#else
  *reinterpret_cast<float4*>(dst) = *reinterpret_cast<const float4*>(src);
#endif
}

__device__ __forceinline__ void stage_wait() {
#if USE_ASYNC_LDS
#  if defined(__has_builtin) && __has_builtin(__builtin_amdgcn_s_wait_asynccnt)
  __builtin_amdgcn_s_wait_asynccnt(0);
#  else
  asm volatile("s_wait_asynccnt 0x0" ::: "memory");
#  endif
#endif
}

__device__ __forceinline__ v8f wmma4(v2f a, v2f b, v8f c) {
  // V_WMMA_F32_16X16X4_F32 : D = A(16x4) x B(4x16) + C(16x16), full fp32
  return __builtin_amdgcn_wmma_f32_16x16x4_f32(
      false, a, false, b, (short)0, c, false, false);
}

// Branch-free transcendentals (libm tanhf expands to a divergent multi-branch
// sequence).
__device__ __forceinline__ float fast_tanh(float x) {
  const float e = __expf(-2.0f * fabsf(x));
  const float r = __fdividef(1.0f - e, 1.0f + e);
  return copysignf(r, x);
}
__device__ __forceinline__ float sigm(float x) {
  return __fdividef(1.0f, 1.0f + __expf(-x));
}

// ---------------------------------------------------------------------------
// Phase 1 (fully parallel over t): attention + pooling.
//   T1 = tanh(H[:,t,:] @ Wa + ba); logits = T1 @ Va; beta = softmax(logits)
//   pooled[t,b,:] = beta[b,:] @ H[b,:,:]
// One wave per (t, 16-row batch tile): 128 * 32 = 4096 waves.
// ---------------------------------------------------------------------------
__global__ __launch_bounds__(128, 1)
void attn_pool_kernel(const float* __restrict__ H, const float* __restrict__ Wa,
                      const float* __restrict__ ba, const float* __restrict__ Va,
                      float* __restrict__ pooled) {
  __shared__ float lds[4][16 * 128];   // per-wave 16x128 staging tile (32 KB)
  const int wave = threadIdx.x >> 5;
  const int lane = threadIdx.x & 31;
  const int ln   = lane & 15;
  const int half = lane >> 4;
  const int gw   = blockIdx.x * 4 + wave;
  const int t    = gw >> 5;
  const int b0   = (gw & 31) << 4;
  float* tile = lds[wave];

  // ---- GEMM1: T1 = tanh(H_t @ Wa + ba), 16x128 tile via f32 WMMA ----
  v8f acc[8] = {};
  const float* Arow = H + ((size_t)(b0 + ln) * S + t) * D;  // H[b, t, :]
  for (int k = 0; k < D; k += 4) {
    v2f a;
    a.x = Arow[k + half * 2 + 0];
    a.y = Arow[k + half * 2 + 1];
#pragma unroll
    for (int n = 0; n < 8; ++n) {
      const int col = n * 16 + ln;
      v2f bf;
      bf.x = Wa[(k + half    ) * S + col];
      bf.y = Wa[(k + half + 2) * S + col];
      acc[n] = wmma4(a, bf, acc[n]);
    }
  }
#pragma unroll
  for (int n = 0; n < 8; ++n) {
    const int col = n * 16 + ln;
    const float bav = ba[col];
#pragma unroll
    for (int r = 0; r < 8; ++r) {
      const int row = r + half * 8;   // C layout: half selects rows 0-7 / 8-15
      tile[row * 128 + col] = fast_tanh(acc[n][r] + bav);
    }
  }
  __syncthreads();

  // ---- GEMM2: logits = T1 @ Va ----
  v8f lg[8] = {};
  for (int k = 0; k < S; k += 4) {
    v2f a;
    a.x = tile[ln * 128 + k + half * 2 + 0];
    a.y = tile[ln * 128 + k + half * 2 + 1];
#pragma unroll
    for (int n = 0; n < 8; ++n) {
      const int col = n * 16 + ln;
      v2f bf;
      bf.x = Va[(k + half    ) * S + col];
      bf.y = Va[(k + half + 2) * S + col];
      lg[n] = wmma4(a, bf, lg[n]);
    }
  }

  // ---- row-wise softmax (row elements live across 16 lanes of one half) ----
#pragma unroll
  for (int r = 0; r < 8; ++r) {
    float m = lg[0][r];
#pragma unroll
    for (int n = 1; n < 8; ++n) m = fmaxf(m, lg[n][r]);
#pragma unroll
    for (int mask = 1; mask <= 8; mask <<= 1) m = fmaxf(m, __shfl_xor(m, mask, 32));
    float e[8];
    float ssum = 0.0f;
#pragma unroll
    for (int n = 0; n < 8; ++n) { e[n] = __expf(lg[n][r] - m); ssum += e[n]; }
#pragma unroll
    for (int mask = 1; mask <= 8; mask <<= 1) ssum += __shfl_xor(ssum, mask, 32);
    const float inv = __fdividef(1.0f, ssum);
    const int row = r + half * 8;
#pragma unroll
    for (int n = 0; n < 8; ++n) tile[row * 128 + n * 16 + ln] = e[n] * inv;
  }
  __syncthreads();

  // ---- pooled[b,:] = beta[b,:] @ H[b,:,:]  (per-row weighted sum, VALU) ----
  const int d4 = lane * 4;
  for (int b = 0; b < 16; ++b) {
    const float* Hb = H + (size_t)(b0 + b) * S * D;
    float a0 = 0.f, a1 = 0.f, a2 = 0.f, a3 = 0.f;
    for (int s = 0; s < S; ++s) {
      const float w = tile[b * 128 + s];
      const float4 hv = *reinterpret_cast<const float4*>(Hb + (size_t)s * D + d4);
      a0 = fmaf(w, hv.x, a0); a1 = fmaf(w, hv.y, a1);
      a2 = fmaf(w, hv.z, a2); a3 = fmaf(w, hv.w, a3);
    }
    float4 o; o.x = a0; o.y = a1; o.z = a2; o.w = a3;
    *reinterpret_cast<float4*>(pooled + ((size_t)t * B + b0 + b) * D + d4) = o;
  }
}

// ---------------------------------------------------------------------------
// Prep: fold y-recursion into U and bias.
//   Ueff[k,c] = U[k,c] + fc_W[k]*Wy[c];  bias_e[c] = bias[c] + fc_b*Wy[c]
// ---------------------------------------------------------------------------
__global__ void prep_kernel(const float* __restrict__ U, const float* __restrict__ fcW,
                            const float* __restrict__ Wy, const float* __restrict__ biasin,
                            const float* __restrict__ fcb,
                            float* __restrict__ Ueff, float* __restrict__ bias_e) {
  int idx = blockIdx.x * blockDim.x + threadIdx.x;
  const int total = HS * G4;
  for (; idx < total; idx += gridDim.x * blockDim.x) {
    const int k = idx / G4, col = idx - k * G4;
    Ueff[idx] = fmaf(fcW[k], Wy[col], U[idx]);
    if (idx < G4) bias_e[idx] = fmaf(fcb[0], Wy[idx], biasin[idx]);
  }
}

__global__ void init_state_kernel(float* __restrict__ h_ws, float* __restrict__ c_ws) {
  int idx = blockIdx.x * blockDim.x + threadIdx.x;
  const int nh = 2 * B * HS, nc = B * HS;
  for (; idx < nh + nc; idx += gridDim.x * blockDim.x) {
    if (idx < nh) h_ws[idx] = 0.0f;
    else          c_ws[idx - nh] = 0.0f;
  }
}

// ---------------------------------------------------------------------------
// Phase 2 (serial over t): one launch per timestep.
//   gates = [pooled_t | h_prev] @ [W ; Ueff] (+ y0 @ Wy at t==0) + bias
// 8 waves per block share one 16-row batch tile; A (16x384, 24KB) is staged
// once per block into LDS (async global->LDS when available). Each wave owns
// a 16x16 (batch x hs) patch and all 4 gate groups so the cell is local.
// ---------------------------------------------------------------------------
__global__ __launch_bounds__(256, 1)
void lstm_step_kernel(const float* __restrict__ pooled_t, const float* __restrict__ W,
                      const float* __restrict__ Ueff, const float* __restrict__ biasv,
                      const float* __restrict__ Wy, const float* __restrict__ y0,
                      const float* __restrict__ hprev, float* __restrict__ hnext,
                      float* __restrict__ c_ws, float* __restrict__ hidden_out,
                      int t, int use_y0) {
  __shared__ float Atile[16 * ASTRIDE];  // 16 rows x 384 K, stride 388 (24.25 KB)
  const int tid  = threadIdx.x;
  const int wave = tid >> 5;
  const int lane = tid & 31;
  const int ln   = lane & 15;
  const int half = lane >> 4;
  const int b0   = (blockIdx.x >> 1) << 4;                  // shared by all 8 waves
  const int j0   = ((((blockIdx.x & 1) << 3) + wave)) << 4; // per-wave hs tile

  // cooperative stage: A = [pooled_t row | h_prev row], 96 float4 per row
  for (int i = tid; i < 16 * 96; i += 256) {
    const int row = i / 96;
    const int c4  = (i - row * 96) * 4;
    const float* src = (c4 < D)
        ? (pooled_t + (size_t)(b0 + row) * D + c4)
        : (hprev    + (size_t)(b0 + row) * HS + (c4 - D));
    stage16(src, &Atile[row * ASTRIDE + c4]);
  }
  stage_wait();
  __syncthreads();

  v8f acc[4] = {};   // i, f, g, o gate accumulators for this 16x16 patch
  const float* Ar = &Atile[ln * ASTRIDE];

  // K-part 1: pooled_t (K = 128) against W
  for (int k = 0; k < D; k += 4) {
    const float2 av = *reinterpret_cast<const float2*>(Ar + k + half * 2);
    v2f a; a.x = av.x; a.y = av.y;
#pragma unroll
    for (int g = 0; g < 4; ++g) {
      const int col = g * HS + j0 + ln;
      v2f bf;
      bf.x = W[(k + half    ) * G4 + col];
      bf.y = W[(k + half + 2) * G4 + col];
      acc[g] = wmma4(a, bf, acc[g]);
    }
  }
  // K-part 2: h_prev (K = 256) against folded Ueff
  for (int k = 0; k < HS; k += 4) {
    const float2 av = *reinterpret_cast<const float2*>(Ar + D + k + half * 2);
    v2f a; a.x = av.x; a.y = av.y;
#pragma unroll
    for (int g = 0; g < 4; ++g) {
      const int col = g * HS + j0 + ln;
      v2f bf;
      bf.x = Ueff[(k + half    ) * G4 + col];
      bf.y = Ueff[(k + half + 2) * G4 + col];
      acc[g] = wmma4(a, bf, acc[g]);
    }
  }

  const int j = j0 + ln;
  const float bi = biasv[0 * HS + j], bff = biasv[1 * HS + j];
  const float bg = biasv[2 * HS + j], bo  = biasv[3 * HS + j];
  const float wyi = Wy[0 * HS + j], wyf = Wy[1 * HS + j];
  const float wyg = Wy[2 * HS + j], wyo = Wy[3 * HS + j];

#pragma unroll
  for (int r = 0; r < 8; ++r) {
    const int b = b0 + r + half * 8;
    float iv = acc[0][r] + bi;
    float fv = acc[1][r] + bff;
    float gv = acc[2][r] + bg;
    float ov = acc[3][r] + bo;
    if (use_y0) {                        // only t==0: explicit y0 @ Wy term
      const float y = y0[b];
      iv = fmaf(y, wyi, iv); fv = fmaf(y, wyf, fv);
      gv = fmaf(y, wyg, gv); ov = fmaf(y, wyo, ov);
    }
    const float si = sigm(iv), sf = sigm(fv), so = sigm(ov);
    const float tg = fast_tanh(gv);
    const size_t idx = (size_t)b * HS + j;
    const float cn = fmaf(sf, c_ws[idx], si * tg);
    const float hn = so * fast_tanh(cn);
    c_ws[idx]  = cn;
    hnext[idx] = hn;
    hidden_out[(size_t)b * (S * HS) + (size_t)t * HS + j] = hn;
  }
}

// Final y = h_final @ fc_W^T + fc_b   (O == 1)
__global__ void final_y_kernel(const float* __restrict__ hfin, const float* __restrict__ fcW,
                               const float* __restrict__ fcb, float* __restrict__ yout) {
  const int b = blockIdx.x * blockDim.x + threadIdx.x;
  if (b < B) {
    float s = fcb[0];
    const float* hr = hfin + (size_t)b * HS;
    for (int k = 0; k < HS; ++k) s = fmaf(hr[k], fcW[k], s);
    yout[b] = s;
  }
}

extern "C" void kernel_launch(void* const* d_in, const int* in_sizes, int n_in,
                              void* d_out, int out_size, void* d_ws, size_t ws_size,
                              hipStream_t stream) {
  const float* H    = (const float*)d_in[0];
  const float* y0   = (const float*)d_in[1];
  const float* Wa   = (const float*)d_in[2];
  // d_in[3] = Ua: unused — s_t stays zero so s_t @ Ua == 0 identically.
  const float* ba   = (const float*)d_in[4];
  const float* Va   = (const float*)d_in[5];
  const float* W    = (const float*)d_in[6];
  const float* U    = (const float*)d_in[7];
  const float* bias = (const float*)d_in[8];
  const float* Wy   = (const float*)d_in[9];
  const float* fcW  = (const float*)d_in[10];
  const float* fcb  = (const float*)d_in[11];

  float* out = (float*)d_out;             // [B] y_f, then [B,S,HS] hidden_seq
  float* ws  = (float*)d_ws;
  float* pooled = ws;                              // S*B*D      (33.5 MB)
  float* h_ws   = pooled + (size_t)S * B * D;      // 2*B*HS     (double buffer)
  float* c_ws   = h_ws + 2 * (size_t)B * HS;       // B*HS
  float* Ueff   = c_ws + (size_t)B * HS;           // HS*G4
  float* bias_e = Ueff + (size_t)HS * G4;          // G4

  init_state_kernel<<<768, 512, 0, stream>>>(h_ws, c_ws);
  prep_kernel<<<256, 256, 0, stream>>>(U, fcW, Wy, bias, fcb, Ueff, bias_e);
  attn_pool_kernel<<<1024, 128, 0, stream>>>(H, Wa, ba, Va, pooled);

  for (int t = 0; t < S; ++t) {
    const float* hp = h_ws + (size_t)(t & 1) * B * HS;
    float*       hn = h_ws + (size_t)((t + 1) & 1) * B * HS;
    lstm_step_kernel<<<64, 256, 0, stream>>>(
        pooled + (size_t)t * B * D, W, Ueff,
        (t == 0) ? bias : bias_e, Wy, y0,
        hp, hn, c_ws, out + B, t, (t == 0) ? 1 : 0);
  }
  // after 128 steps the final h lands in buffer 0
  final_y_kernel<<<2, 256, 0, stream>>>(h_ws, fcW, fcb, out);
}